// BaseTransformer_31885837206061
// MI455X (gfx1250) — compile-verified
//
#include <hip/hip_runtime.h>
#include <hip/hip_bf16.h>

typedef __attribute__((ext_vector_type(2))) float v2f;
typedef __attribute__((ext_vector_type(8))) float v8f;

#define BB   8
#define EE   16
#define CIN  64
#define CC   64
#define HWH  4096   // 64*64
#define SEGS 8      // K-split for the gram matrix

__device__ __forceinline__ float selu_f(float x) {
    const float alpha = 1.6732632423543772f;
    const float scale = 1.0507009873554805f;
    return x > 0.0f ? scale * x : scale * alpha * (__expf(x) - 1.0f);
}

__device__ __forceinline__ v8f wmma_f32(v2f a, v2f b, v8f c) {
    return __builtin_amdgcn_wmma_f32_16x16x4_f32(false, a, false, b, (short)0, c, false, false);
}

// ---------------------------------------------------------------------------
// Kernel 1: three fused 1x1 convs.  One wave = one 16x16 (c-tile x hw-tile)
// output tile for one (b,e) slice.  A = W[64,64], B = X[64,4096] (K=CIN=64).
// value (no activation) -> d_out;  key/query (selu) -> workspace.
// waves = B*E * 4 (m-tiles) * 256 (n-tiles) = 131072
// ---------------------------------------------------------------------------
__global__ void k_conv3(const float* __restrict__ X,
                        const float* __restrict__ Wv, const float* __restrict__ bv,
                        const float* __restrict__ Wk, const float* __restrict__ bk,
                        const float* __restrict__ Wq, const float* __restrict__ bq,
                        float* __restrict__ value,
                        float* __restrict__ key,
                        float* __restrict__ query)
{
    const int wid   = blockIdx.x * (blockDim.x >> 5) + (threadIdx.x >> 5);
    const int lane  = threadIdx.x & 31;
    const int lhalf = lane >> 4;
    const int lmod  = lane & 15;

    const int ntile = wid & 255;
    const int mt    = (wid >> 8) & 3;
    const int be    = wid >> 10;          // 0..127  (= b*E + e)

    const int m0 = mt * 16;
    const int n0 = ntile * 16;
    const float* Xp = X + (size_t)be * CIN * HWH;

    v8f cv = {}; v8f ck = {}; v8f cq = {};
    const int arow = m0 + lmod;

    for (int k0 = 0; k0 < CIN; k0 += 4) {
        const int ka = k0 + 2 * lhalf;
        v2f av = *(const v2f*)(Wv + arow * CIN + ka);
        v2f ak = *(const v2f*)(Wk + arow * CIN + ka);
        v2f aq = *(const v2f*)(Wq + arow * CIN + ka);
        v2f bf;
        bf.x = Xp[(size_t)ka * HWH + n0 + lmod];
        bf.y = Xp[(size_t)(ka + 1) * HWH + n0 + lmod];
        cv = wmma_f32(av, bf, cv);
        ck = wmma_f32(ak, bf, ck);
        cq = wmma_f32(aq, bf, cq);
    }

    const size_t obase = (size_t)be * CC * HWH;
    for (int r = 0; r < 8; ++r) {
        const int m = m0 + r + 8 * lhalf;
        const size_t oi = obase + (size_t)m * HWH + n0 + lmod;
        value[oi] = cv[r] + bv[m];
        key[oi]   = selu_f(ck[r] + bk[m]);
        query[oi] = selu_f(cq[r] + bq[m]);
    }
}

// ---------------------------------------------------------------------------
// Kernel 2: partial gram:  part[b,c,seg,i,j] = sum_{hw in seg} key_i * query_j
// One wave per (b, c, seg), SEGS segments of HWH/SEGS -> 128 WMMAs each.
// Each wave owns a private 256-float partial tile: no atomics, deterministic.
// A-lane row (M=i) = lane%16 on key; B-lane col (N=j) = lane%16 on query:
// both fragment loads are contiguous v2f in hw (fully coalesced).
// waves = 8*64*8 = 4096
// ---------------------------------------------------------------------------
__global__ void k_dots(const float* __restrict__ key,
                       const float* __restrict__ query,
                       float* __restrict__ part)
{
    const int wid   = blockIdx.x * (blockDim.x >> 5) + (threadIdx.x >> 5);
    const int lane  = threadIdx.x & 31;
    const int lhalf = lane >> 4;
    const int lmod  = lane & 15;

    const int seg = wid & (SEGS - 1);
    const int c   = (wid >> 3) & 63;
    const int b   = wid >> 9;

    const size_t rbase = ((size_t)(b * EE + lmod) * CC + c) * HWH;  // row = ensemble member lmod
    const int hw0 = seg * (HWH / SEGS);

    v8f acc = {};
    for (int k0 = hw0; k0 < hw0 + (HWH / SEGS); k0 += 4) {
        const int kk = k0 + 2 * lhalf;
        v2f a  = *(const v2f*)(key   + rbase + kk);
        v2f bf = *(const v2f*)(query + rbase + kk);
        acc = wmma_f32(a, bf, acc);
    }

    float* dp = part + ((size_t)(b * CC + c) * SEGS + seg) * 256;
    for (int r = 0; r < 8; ++r) {
        const int i = r + 8 * lhalf;        // key index (rows)
        dp[i * 16 + lmod] = acc[r];
    }
}

// ---------------------------------------------------------------------------
// Kernel 3: deterministic partial reduction + softmax over i (axis -2).
// one thread per (b,c,j) = 8192 threads; writes weights[b,c,i,j].
// ---------------------------------------------------------------------------
__global__ void k_softmax(const float* __restrict__ part,
                          float* __restrict__ weights)
{
    const int idx = blockIdx.x * blockDim.x + threadIdx.x;
    if (idx >= BB * CC * EE) return;
    const int j  = idx & 15;
    const int bc = idx >> 4;
    const float* pp = part + (size_t)bc * SEGS * 256 + j;

    float d[16];
    for (int i = 0; i < 16; ++i) {
        float s = 0.0f;
        for (int seg = 0; seg < SEGS; ++seg) s += pp[seg * 256 + i * 16];
        d[i] = s;
    }
    float mx = -3.402823466e38f;
    for (int i = 0; i < 16; ++i) mx = fmaxf(mx, d[i]);
    float s = 0.0f;
    for (int i = 0; i < 16; ++i) { d[i] = __expf(d[i] - mx); s += d[i]; }
    const float inv = 1.0f / s;
    float* wo = weights + (size_t)bc * 256 + j;
    for (int i = 0; i < 16; ++i) wo[i * 16] = d[i] * inv;
}

// ---------------------------------------------------------------------------
// Kernel 4: mean over ensemble:  mean[b,c,hw] = (1/16) sum_i value[b,i,c,hw]
// one thread per (b,c,hw) = 2M threads, coalesced over hw.
// ---------------------------------------------------------------------------
__global__ void k_mean(const float* __restrict__ value, float* __restrict__ mean)
{
    const int idx = blockIdx.x * blockDim.x + threadIdx.x;
    if (idx >= BB * CC * HWH) return;
    const int hw = idx & (HWH - 1);
    const int c  = (idx >> 12) & 63;
    const int b  = idx >> 18;
    const size_t base = ((size_t)b * EE * CC + c) * HWH + hw;
    const size_t istr = (size_t)CC * HWH;
    float s = 0.0f;
    for (int i = 0; i < 16; ++i) s += value[base + (size_t)i * istr];
    mean[idx] = s * 0.0625f;
}

// ---------------------------------------------------------------------------
// Kernel 5: out[b,j,c,hw] = selu(mean + sum_i weights[b,c,i,j]*(value[b,i,c,hw]-mean))
// One wave per (b, c, 256-wide hw block).  A = weights^T loaded once (4 chunks),
// then 16 N-tiles x 4 WMMAs.  In-place on d_out: each wave reads all i and
// writes all j for a private set of hw columns (disjoint across waves).
// waves = 8*64*16 = 8192
// ---------------------------------------------------------------------------
__global__ void k_combine(const float* __restrict__ weights,
                          const float* __restrict__ mean,
                          float* __restrict__ value /* == d_out, in/out */)
{
    const int wid   = blockIdx.x * (blockDim.x >> 5) + (threadIdx.x >> 5);
    const int lane  = threadIdx.x & 31;
    const int lhalf = lane >> 4;
    const int lmod  = lane & 15;

    const int nblk = wid & 15;
    const int c    = (wid >> 4) & 63;
    const int b    = wid >> 10;

    // A fragments: A[m=j][k=i] = weights[b,c,i,j]  (transpose of ij)
    const float* wp = weights + (size_t)(b * CC + c) * 256;
    v2f afr[4];
    for (int ch = 0; ch < 4; ++ch) {
        const int i = 4 * ch + 2 * lhalf;
        afr[ch].x = wp[i * 16 + lmod];
        afr[ch].y = wp[(i + 1) * 16 + lmod];
    }

    const size_t vbase = ((size_t)b * EE * CC + c) * HWH;
    const size_t istr  = (size_t)CC * HWH;
    const float* mp    = mean + (size_t)(b * CC + c) * HWH;

    for (int t = 0; t < 16; ++t) {
        const int hw = nblk * 256 + t * 16 + lmod;
        const float mv = mp[hw];
        v8f acc = {};
        for (int ch = 0; ch < 4; ++ch) {
            const int i = 4 * ch + 2 * lhalf;
            v2f bf;
            bf.x = value[vbase + (size_t)i * istr + hw] - mv;
            bf.y = value[vbase + (size_t)(i + 1) * istr + hw] - mv;
            acc = wmma_f32(afr[ch], bf, acc);
        }
        for (int r = 0; r < 8; ++r) {
            const int j = r + 8 * lhalf;
            value[vbase + (size_t)j * istr + hw] = selu_f(mv + acc[r]);
        }
    }
}

// ---------------------------------------------------------------------------
extern "C" void kernel_launch(void* const* d_in, const int* in_sizes, int n_in,
                              void* d_out, int out_size, void* d_ws, size_t ws_size,
                              hipStream_t stream)
{
    const float* X  = (const float*)d_in[0];
    const float* Wv = (const float*)d_in[1];
    const float* bv = (const float*)d_in[2];
    const float* Wk = (const float*)d_in[3];
    const float* bk = (const float*)d_in[4];
    const float* Wq = (const float*)d_in[5];
    const float* bq = (const float*)d_in[6];
    float* out = (float*)d_out;                     // value, then final output (in place)

    const size_t TEN = (size_t)BB * EE * CC * HWH;  // 33,554,432 floats
    float* ws      = (float*)d_ws;
    float* key     = ws;                                        // 128 MiB
    float* query   = ws + TEN;                                  // 128 MiB
    float* part    = ws + 2 * TEN;                              // B*C*SEGS*256 = 4 MiB
    float* weights = part + (size_t)BB * CC * SEGS * 256;       // B*C*256 = 512 KiB
    float* meanb   = weights + (size_t)BB * CC * 256;           // B*C*HW  = 8 MiB

    // 1) fused V/K/Q 1x1 convs (value -> d_out)
    k_conv3<<<16384, 256, 0, stream>>>(X, Wv, bv, Wk, bk, Wq, bq, out, key, query);

    // 2) gram matrix partials (deterministic, no atomics)
    k_dots<<<512, 256, 0, stream>>>(key, query, part);

    // 3) partial reduction + softmax over key axis -> weights
    k_softmax<<<32, 256, 0, stream>>>(part, weights);

    // 4) ensemble mean of value
    k_mean<<<8192, 256, 0, stream>>>(out, meanb);

    // 5) weighted perturbation combine + selu (in place on d_out)
    k_combine<<<1024, 256, 0, stream>>>(weights, meanb, out);
}